// PolyMatchingLoss_13554916786703
// MI455X (gfx1250) — compile-verified
//
#include <hip/hip_runtime.h>
#include <math.h>

#define PNUM  1024
#define BATCH 128

typedef __attribute__((ext_vector_type(2))) float v2f;
typedef __attribute__((ext_vector_type(8))) float v8f;

// smooth_l1(dx) + smooth_l1(dy), branchless:
//  w = min(|d|,1);  sl1 = w*(|d| - 0.5*w)   [ =0.5d^2 if |d|<1, |d|-0.5 else ]
__device__ __forceinline__ float sl1_pair(float dx, float dy) {
    float ax = fabsf(dx);
    float ay = fabsf(dy);
    float wx = fminf(ax, 1.0f);
    float wy = fminf(ay, 1.0f);
    float tx = fmaf(-0.5f, wx, ax);
    float ty = fmaf(-0.5f, wy, ay);
    return fmaf(wx, tx, wy * ty);
}

// One block per batch. 512 threads = 16 waves (wave32).
// Wave w handles shift tiles baseS = (w*4 + k)*16, k=0..3  -> 16*4*16 = 1024 shifts.
// Within a tile, lane l covers shift M = l&15 and K-slots {0,1} (lanes 0-15) or
// {2,3} (lanes 16-31): exactly the 16x4 f32 A-matrix VGPR layout.
// D(16x16) = A(16x4) * ones(4x16) + C accumulates per-shift partial sums.
__global__ __launch_bounds__(512)
void poly_match_min_kernel(const float* __restrict__ pred,
                           const float* __restrict__ gt,
                           float* __restrict__ minOut) {
    __shared__ float4 sPred4[PNUM / 2];   //  8 KB: pred[b], point pairs
    __shared__ float4 sGt4[PNUM];         // 16 KB: gt[b] duplicated (no modulo)
    __shared__ float  sRed[16];

    const int b    = blockIdx.x;
    const int tid  = threadIdx.x;
    const int lane = tid & 31;
    const int wave = tid >> 5;

    // ---- stage into LDS (whole batch row; reused 1024x afterwards) ----
    const float4* pg = (const float4*)(pred + (size_t)b * (PNUM * 2));
    const float4* gg = (const float4*)(gt   + (size_t)b * (PNUM * 2));
    float4 pv = pg[tid];
    float4 gv = gg[tid];
    sPred4[tid] = pv;
    sGt4[tid] = gv;
    sGt4[tid + PNUM / 2] = gv;            // duplicate: gt[i] == gt[i+1024]
    __syncthreads();

    const float2* sGt2 = (const float2*)sGt4;

    const int m    = lane & 15;           // shift within tile (A-matrix M)
    const int koff = (lane >> 4) << 1;    // 0 or 2 (A-matrix K half)
    const v2f bOnes = (v2f){1.0f, 1.0f};  // B = all-ones 4x16

    float waveMin = 3.0e38f;

    for (int k = 0; k < 4; ++k) {
        const int baseS = (wave * 4 + k) << 4;
        const int s = baseS + m;
        v8f c = {};
        #pragma unroll 4
        for (int jb = 0; jb < PNUM; jb += 4) {
            const int j = jb + koff;                  // even
            const float4 p  = sPred4[j >> 1];         // pred[j], pred[j+1] (b128)
            const int    e  = s + j;                  // <= 2046
            const float2 g0 = sGt2[e];                // gt[s+j]     (b64)
            const float2 g1 = sGt2[e + 1];            // gt[s+j+1]   (b64)
            float cost0 = sl1_pair(p.x - g0.x, p.y - g0.y);   // K = koff
            float cost1 = sl1_pair(p.z - g1.x, p.w - g1.y);   // K = koff+1
            v2f a = (v2f){cost0, cost1};
            // D = A(16x4) * ones(4x16) + C : per-shift sums replicated over N
            c = __builtin_amdgcn_wmma_f32_16x16x4_f32(
                    false, a, false, bOnes, (short)0, c, false, false);
        }
        // lane holds shifts baseS + 8*(lane>=16) + r in c[r], r=0..7
        float v = c[0];
        v = fminf(v, c[1]); v = fminf(v, c[2]); v = fminf(v, c[3]);
        v = fminf(v, c[4]); v = fminf(v, c[5]); v = fminf(v, c[6]);
        v = fminf(v, c[7]);
        // wave-wide min -> min over this 16-shift tile
        for (int off = 16; off > 0; off >>= 1)
            v = fminf(v, __shfl_xor(v, off, 32));
        waveMin = fminf(waveMin, v);
    }

    if (lane == 0) sRed[wave] = waveMin;
    __syncthreads();
    if (wave == 0) {
        float v = (lane < 16) ? sRed[lane] : 3.0e38f;
        for (int off = 16; off > 0; off >>= 1)
            v = fminf(v, __shfl_xor(v, off, 32));
        if (lane == 0) minOut[b] = v * (1.0f / (float)PNUM);  // mean over j
    }
}

// Single wave: mean of the 128 per-batch minima.
__global__ __launch_bounds__(32)
void poly_match_mean_kernel(const float* __restrict__ minOut,
                            float* __restrict__ out) {
    const int lane = threadIdx.x;
    float v = minOut[lane] + minOut[lane + 32] +
              minOut[lane + 64] + minOut[lane + 96];
    for (int off = 16; off > 0; off >>= 1)
        v += __shfl_xor(v, off, 32);
    if (lane == 0) out[0] = v * (1.0f / (float)BATCH);
}

extern "C" void kernel_launch(void* const* d_in, const int* in_sizes, int n_in,
                              void* d_out, int out_size, void* d_ws, size_t ws_size,
                              hipStream_t stream) {
    const float* pred = (const float*)d_in[0];   // (128, 1024, 2) f32
    const float* gt   = (const float*)d_in[1];   // (128, 1024, 2) f32
    float* out   = (float*)d_out;                // scalar f32
    float* minWs = (float*)d_ws;                 // 128 floats of scratch

    poly_match_min_kernel<<<dim3(BATCH), dim3(512), 0, stream>>>(pred, gt, minWs);
    poly_match_mean_kernel<<<dim3(1), dim3(32), 0, stream>>>(minWs, out);
}